// FinePreprocess_63221918597660
// MI455X (gfx1250) — compile-verified
//
#include <hip/hip_runtime.h>
#include <cstdint>

// Problem constants (from reference setup_inputs)
#define C_CH   128
#define H_IN   240
#define W_IN   320
#define STRIDE 4
#define LW     80      // windows per row: (320 + 2*pad - w)/4 + 1 == 80 for both cases
#define M_WIN  4096

// Native clang vector type (HIP's float4 is a class; builtins need real vectors)
typedef float v4f __attribute__((ext_vector_type(4)));

// One block gathers one window patch.
//   KW   = window size (5 or 9)
//   PADT = padding (2 or 4)
// Output layout per window: (p = kh*KW+kw, c)  -> w*w*128 contiguous floats.
template <int KW, int PADT>
__global__ __launch_bounds__(256) void gather_windows_kernel(
    const float* __restrict__ x,
    const int* __restrict__ b_idx,
    const int* __restrict__ l_idx,
    float* __restrict__ out)
{
    constexpr int PP  = KW * KW;        // positions per window
    constexpr int ROW = C_CH + 4;       // LDS row stride in floats: 528 B = 16-B aligned,
                                        // bank stride 4 -> at most 2-way conflicts
    __shared__ __align__(16) float tile[PP * ROW];

    const int m   = blockIdx.x;
    const int tid = threadIdx.x;

    const int b  = b_idx[m];
    const int l  = l_idx[m];
    const int ih = l / LW;
    const int iw = l - ih * LW;
    const int gh0 = ih * STRIDE - PADT;
    const int gw0 = iw * STRIDE - PADT;

    // ---- Phase 0: zero LDS tile (provides the zero padding for OOB taps) ----
    for (int i = tid; i < PP * ROW; i += 256) tile[i] = 0.0f;
    __syncthreads();

    // ---- Phase 1: async gather global -> LDS with on-the-fly transpose ----
    // Enumerate elements input-major (c slow; kh, kw fast) so a wave's global
    // reads are runs of KW contiguous floats. Base address (uniform) rides in
    // an SGPR pair (GVS mode); per-lane part is a 32-bit byte offset, so the
    // per-element VALU cost is one 32-bit mad + shift instead of 64-bit adds.
    const unsigned long long base =
        (unsigned long long)(x + (size_t)b * C_CH * H_IN * W_IN);
    for (int e = tid; e < C_CH * PP; e += 256) {
        const int c  = e / PP;
        const int p  = e - c * PP;
        const int kh = p / KW;
        const int kw = p - kh * KW;
        const int gh = gh0 + kh;
        const int gw = gw0 + kw;
        if ((unsigned)gh < (unsigned)H_IN && (unsigned)gw < (unsigned)W_IN) {
            const unsigned voff =
                (unsigned)((c * (H_IN * W_IN) + gh * W_IN + gw) * 4);
            const unsigned ldsaddr = (unsigned)(size_t)(&tile[p * ROW + c]);
            // copies 4 bytes from global address base + voff to the LDS address in ldsaddr  (ASYNCcnt-tracked, EXEC-masked)
            asm volatile("global_load_async_to_lds_b32 %0, %1, %2"
                         :: "v"(ldsaddr), "v"(voff), "s"(base)
                         : "memory");
        }
    }
    // Wait for this wave's async transfers, then sync the workgroup.
    asm volatile("s_wait_asynccnt 0" ::: "memory");
    __syncthreads();

    // ---- Phase 2: coalesced 128-bit non-temporal stores, output-major ----
    // Output is written once and never re-read: NT keeps L2 for input reuse.
    v4f* out4 = (v4f*)(out + (size_t)m * PP * C_CH);
    constexpr int N4 = PP * C_CH / 4;
    for (int e4 = tid; e4 < N4; e4 += 256) {
        const int p  = e4 / (C_CH / 4);
        const int c4 = e4 - p * (C_CH / 4);
        const v4f v = *(const v4f*)&tile[p * ROW + c4 * 4];
        __builtin_nontemporal_store(v, &out4[e4]);  // global_store_b128 th:NT
    }
}

extern "C" void kernel_launch(void* const* d_in, const int* in_sizes, int n_in,
                              void* d_out, int out_size, void* d_ws, size_t ws_size,
                              hipStream_t stream) {
    (void)in_sizes; (void)n_in; (void)out_size; (void)d_ws; (void)ws_size;

    const float* x0    = (const float*)d_in[0];
    const float* x1    = (const float*)d_in[1];
    const int*   b_idx = (const int*)d_in[2];
    const int*   i_idx = (const int*)d_in[3];
    const int*   j_idx = (const int*)d_in[4];

    float* out0 = (float*)d_out;
    float* out1 = out0 + (size_t)M_WIN * 25 * C_CH;   // out0 is (4096, 25, 128)

    // out0: w=5, pad=2 ; out1: w=9, pad=4
    gather_windows_kernel<5, 2><<<M_WIN, 256, 0, stream>>>(x0, b_idx, i_idx, out0);
    gather_windows_kernel<9, 4><<<M_WIN, 256, 0, stream>>>(x1, b_idx, j_idx, out1);
}